// SSIMLoss_52192442581415
// MI455X (gfx1250) — compile-verified
//
#include <hip/hip_runtime.h>

// ---------------------------------------------------------------------------
// Fused SSIM loss for MI455X (gfx1250, wave32).
//   - one pass over the inputs (memory-bound problem: 134 MB @ 23.3 TB/s)
//   - horizontal 11-tap box sums via running sums in LDS (5 quantities)
//   - vertical 11-tap box sums via V_WMMA_F32_16X16X4_F32 banded matmul
//   - pointwise SSIM + block reduce + f64 global atomic, then finalize.
// ---------------------------------------------------------------------------

#define TILE 32          // output tile per workgroup (32x32)
#define EXT  42          // TILE + 2*5 halo
#define INS  43          // sIn row stride (gcd(43,64)=1 -> conflict-free)
#define HS   41          // sH  row stride (gcd(41,64)=1 -> conflict-free)
#define N_TOTAL 16777216.0  // 64*512*512

typedef float v2f __attribute__((ext_vector_type(2)));
typedef float v8f __attribute__((ext_vector_type(8)));

__global__ void ssim_zero_kernel(double* acc) { acc[0] = 0.0; }

__device__ __forceinline__ float qmake(int q, float a, float b) {
  return (q == 0) ? a : (q == 1) ? b : (q == 2) ? a * a : (q == 3) ? b * b : a * b;
}

__global__ __launch_bounds__(128) void ssim_tile_kernel(
    const float* __restrict__ img1, const float* __restrict__ img2,
    double* __restrict__ acc)
{
  __shared__ float sIn1[EXT * INS];       // 7,224 B
  __shared__ float sIn2[EXT * INS];       // 7,224 B
  __shared__ float sH[5 * EXT * HS];      // 34,440 B : 5 horizontal box-sum maps
  __shared__ float sRed[128];

  const int tid = threadIdx.x;
  const int oy  = blockIdx.y * TILE;
  const int ox  = blockIdx.x * TILE;
  const size_t base = (size_t)blockIdx.z * (512 * 512);

  // ---- stage 42x42 halo region (zero padding outside the image) ----------
  for (int i = tid; i < EXT * EXT; i += 128) {
    int r = i / EXT, c = i % EXT;
    int gy = oy - 5 + r, gx = ox - 5 + c;
    bool ok = (gy >= 0) & (gy < 512) & (gx >= 0) & (gx < 512);
    size_t idx = base + (size_t)gy * 512 + (size_t)gx;
    sIn1[r * INS + c] = ok ? img1[idx] : 0.0f;
    sIn2[r * INS + c] = ok ? img2[idx] : 0.0f;
  }
  __syncthreads();

  // ---- horizontal 11-tap box sums via running sums (5 quantities) --------
  // task t = (q, row); 5*42 = 210 tasks over 128 threads.
  for (int t = tid; t < 5 * EXT; t += 128) {
    int q = t / EXT, r = t % EXT;
    const float* p1 = &sIn1[r * INS];
    const float* p2 = &sIn2[r * INS];
    float s = 0.0f;
#pragma unroll
    for (int j = 0; j < 11; ++j) s += qmake(q, p1[j], p2[j]);
    float* hrow = &sH[(q * EXT + r) * HS];
    hrow[0] = s;
    for (int c = 1; c < TILE; ++c) {
      s += qmake(q, p1[c + 10], p2[c + 10]) - qmake(q, p1[c - 1], p2[c - 1]);
      hrow[c] = s;
    }
  }
  __syncthreads();

  // ---- vertical 11-tap box sums as banded matmul on the WMMA pipe --------
  // Each wave owns one 16x16 subtile: D = Band(16x28) x Hsum(28x16),
  // chained as 7 x V_WMMA_F32_16X16X4_F32 per quantity.
  const int wave = tid >> 5;
  const int lane = tid & 31;
  const int m    = lane & 15;          // A-matrix row (output row in subtile)
  const int kh2  = (lane >> 4) << 1;   // lane-half K offset: 0 or 2
  const int sy   = (wave >> 1) << 4;   // subtile origin inside 32x32 tile
  const int sx   = (wave & 1) << 4;
  const int col  = sx + m;             // B-matrix column = lane&15

  v8f accv[5] = {v8f{}, v8f{}, v8f{}, v8f{}, v8f{}};

#pragma unroll
  for (int kc = 0; kc < 7; ++kc) {
    const int k0 = kc * 4 + kh2;       // this lane-half's K indices: k0, k0+1
    v2f av;
    // band: output row m sums extended rows ir in [m, m+10]
    av[0] = (k0     >= m && k0     <= m + 10) ? 1.0f : 0.0f;
    av[1] = (k0 + 1 >= m && k0 + 1 <= m + 10) ? 1.0f : 0.0f;
    int r0 = sy + k0;     if (r0 > EXT - 1) r0 = EXT - 1;   // pad K (A==0 there)
    int r1 = sy + k0 + 1; if (r1 > EXT - 1) r1 = EXT - 1;
#pragma unroll
    for (int q = 0; q < 5; ++q) {
      v2f bv;
      bv[0] = sH[(q * EXT + r0) * HS + col];
      bv[1] = sH[(q * EXT + r1) * HS + col];
      accv[q] = __builtin_amdgcn_wmma_f32_16x16x4_f32(
          /*neg_a=*/false, av, /*neg_b=*/false, bv,
          /*c_mod=*/(short)0, accv[q], /*reuse_a=*/false, /*reuse_b=*/false);
    }
  }

  // ---- pointwise SSIM over this lane's 8 accumulator elements ------------
  const float inv121 = 1.0f / 121.0f;
  const float C1 = 1e-4f;    // 0.01^2
  const float C2 = 9e-4f;    // 0.03^2
  float local = 0.0f;
#pragma unroll
  for (int e = 0; e < 8; ++e) {
    float mu1 = accv[0][e] * inv121;
    float mu2 = accv[1][e] * inv121;
    float mu1sq = mu1 * mu1, mu2sq = mu2 * mu2, mu12 = mu1 * mu2;
    float s1  = accv[2][e] * inv121 - mu1sq;
    float s2  = accv[3][e] * inv121 - mu2sq;
    float s12 = accv[4][e] * inv121 - mu12;
    float num = (2.0f * mu12 + C1) * (2.0f * s12 + C2);
    float den = (mu1sq + mu2sq + C1) * (s1 + s2 + C2);
    local += num / den;
  }

  // ---- block reduction + one f64 atomic per workgroup --------------------
  sRed[tid] = local;
  __syncthreads();
  for (int s = 64; s > 0; s >>= 1) {
    if (tid < s) sRed[tid] += sRed[tid + s];
    __syncthreads();
  }
  if (tid == 0) atomicAdd(acc, (double)sRed[0]);
}

__global__ void ssim_final_kernel(const double* __restrict__ acc,
                                  float* __restrict__ out) {
  out[0] = (float)(1.0 - acc[0] / N_TOTAL);
}

extern "C" void kernel_launch(void* const* d_in, const int* in_sizes, int n_in,
                              void* d_out, int out_size, void* d_ws, size_t ws_size,
                              hipStream_t stream) {
  (void)in_sizes; (void)n_in; (void)out_size; (void)ws_size;
  const float* img1 = (const float*)d_in[0];
  const float* img2 = (const float*)d_in[1];
  double* acc = (double*)d_ws;
  float*  out = (float*)d_out;

  hipLaunchKernelGGL(ssim_zero_kernel, dim3(1), dim3(1), 0, stream, acc);

  dim3 grid(512 / TILE, 512 / TILE, 64);   // 16 x 16 x 64 tiles
  hipLaunchKernelGGL(ssim_tile_kernel, grid, dim3(128), 0, stream,
                     img1, img2, acc);

  hipLaunchKernelGGL(ssim_final_kernel, dim3(1), dim3(1), 0, stream, acc, out);
}